// Hybrid_ToMe_Masked_Modeling_11458972745812
// MI455X (gfx1250) — compile-verified
//
#include <hip/hip_runtime.h>

// Problem constants (fixed by the reference).
#define BB 16
#define TT 4096
#define CC 768
#define RR 1024
#define HT 2048          // T/2
#define OUTT 3072        // (HT - R) + HT tokens per batch in output

typedef float v2f __attribute__((ext_vector_type(2)));
typedef float v8f __attribute__((ext_vector_type(8)));

// ---------------------------------------------------------------------------
// 1) inv_norm[b,t] = 1/||metric[b,t,:]||  (one wave32 per row)
// ---------------------------------------------------------------------------
__global__ __launch_bounds__(256) void k_inv_norm(const float* __restrict__ metric,
                                                  float* __restrict__ inv_norm) {
    int wave = (int)((blockIdx.x * blockDim.x + threadIdx.x) >> 5);
    int lane = (int)(threadIdx.x & 31);
    if (wave >= BB * TT) return;
    const float* row = metric + (size_t)wave * CC;
    float s = 0.f;
    for (int k = lane; k < CC; k += 32) { float v = row[k]; s += v * v; }
#pragma unroll
    for (int m = 16; m >= 1; m >>= 1) s += __shfl_xor(s, m, 32);
    if (lane == 0) inv_norm[wave] = rsqrtf(s);
}

// ---------------------------------------------------------------------------
// 2) Fused scores GEMM (full-precision f32 WMMA 16x16x4) + row max/argmax.
//    scores must stay f32-faithful: they feed an argsort whose adjacent gaps
//    (~2.5e-5) are far below bf16 noise (~2e-4).
//    Block: 128 threads (4 waves); each wave owns a 16-row strip and walks
//    two 16-col N-tiles at a time (shared A fragment, 2 accumulators).
// ---------------------------------------------------------------------------
__global__ __launch_bounds__(128) void k_scores_argmax(const float* __restrict__ metric,
                                                       const float* __restrict__ inv_norm,
                                                       float* __restrict__ node_max,
                                                       int*   __restrict__ node_idx) {
    const int b      = (int)blockIdx.y;
    const int wave   = (int)(threadIdx.x >> 5);
    const int lane   = (int)(threadIdx.x & 31);
    const int half   = lane >> 4;
    const int lid    = lane & 15;
    const int m_base = (int)blockIdx.x * 64 + wave * 16;   // a-row strip base

    const float* mb   = metric   + (size_t)b * TT * CC;
    const float* invb = inv_norm + (size_t)b * TT;

    // A row feeding this lane (M = lid for both halves).
    const int arow   = m_base + lid;                       // 0..2047
    const int agrow  = 2 * arow;                           // even token
    const float ainv = invb[agrow];
    const float* ap  = mb + (size_t)agrow * CC;

    float rmax[8];
    int   ridx[8];
#pragma unroll
    for (int r = 0; r < 8; ++r) { rmax[r] = -INFINITY; ridx[r] = 0x7fffffff; }

    for (int n0 = 0; n0 < HT; n0 += 32) {
        const int col0 = n0 + lid;                         // first N-tile column
        const int col1 = n0 + 16 + lid;                    // second N-tile column
        const int bg0  = 2 * col0 + 1;                     // odd tokens
        const int bg1  = 2 * col1 + 1;
        const float binv0 = invb[bg0];
        const float binv1 = invb[bg1];
        const float* bp0  = mb + (size_t)bg0 * CC;
        const float* bp1  = mb + (size_t)bg1 * CC;

        v8f acc0 = {};
        v8f acc1 = {};
#pragma unroll 4
        for (int k0 = 0; k0 < CC; k0 += 4) {
            const int kk = k0 + 2 * half;                  // lane's K pair
            // A fragment (16x4 f32): VGPR v holds K = 2*half + v.
            v2f afrag; afrag[0] = ap[kk] * ainv; afrag[1] = ap[kk + 1] * ainv;
            // B fragments (4x16 f32): VGPR v holds K = 2*half + v, N = lid.
            v2f bfrag0; bfrag0[0] = bp0[kk] * binv0; bfrag0[1] = bp0[kk + 1] * binv0;
            v2f bfrag1; bfrag1[0] = bp1[kk] * binv1; bfrag1[1] = bp1[kk + 1] * binv1;

            acc0 = __builtin_amdgcn_wmma_f32_16x16x4_f32(
                false, afrag, false, bfrag0, (short)0, acc0, false, false);
            acc1 = __builtin_amdgcn_wmma_f32_16x16x4_f32(
                false, afrag, false, bfrag1, (short)0, acc1, false, false);
        }
        // Running per-lane max: acc[r] is (row = r + 8*half, col). Update col0
        // first, strict '>' keeps the earliest column (argmax first-hit).
#pragma unroll
        for (int r = 0; r < 8; ++r) {
            if (acc0[r] > rmax[r]) { rmax[r] = acc0[r]; ridx[r] = col0; }
            if (acc1[r] > rmax[r]) { rmax[r] = acc1[r]; ridx[r] = col1; }
        }
    }

    // Cross-lane argmax within each 16-lane half (ties -> lower column).
#pragma unroll
    for (int r = 0; r < 8; ++r) {
#pragma unroll
        for (int m = 1; m <= 8; m <<= 1) {
            float om = __shfl_xor(rmax[r], m, 32);
            int   oi = __shfl_xor(ridx[r], m, 32);
            if (om > rmax[r] || (om == rmax[r] && oi < ridx[r])) { rmax[r] = om; ridx[r] = oi; }
        }
    }
    if (lid == 0) {
#pragma unroll
        for (int r = 0; r < 8; ++r) {
            int row = m_base + half * 8 + r;
            node_max[(size_t)b * HT + row] = rmax[r];
            node_idx[(size_t)b * HT + row] = ridx[r];
        }
    }
}

// ---------------------------------------------------------------------------
// 3) Stable descending rank-sort of node_max (2048 keys/batch, O(n^2)).
//    edge_idx[b][rank] = i  <=>  jnp.argsort(-node_max)
// ---------------------------------------------------------------------------
__global__ __launch_bounds__(256) void k_rank(const float* __restrict__ node_max,
                                              int* __restrict__ edge_idx) {
    __shared__ float v[HT];
    const int b = (int)blockIdx.x;
    const float* nm = node_max + (size_t)b * HT;
    for (int i = (int)threadIdx.x; i < HT; i += 256) v[i] = nm[i];
    __syncthreads();
    for (int i = (int)threadIdx.x; i < HT; i += 256) {
        const float vi = v[i];
        int rank = 0;
        for (int j = 0; j < HT; ++j) {
            float vj = v[j];
            rank += (vj > vi) || (vj == vi && j < i);
        }
        edge_idx[(size_t)b * HT + rank] = i;
    }
}

// ---------------------------------------------------------------------------
// 4) Initialize output rows: unm rows gather src, dst rows copy odd tokens.
// ---------------------------------------------------------------------------
__global__ __launch_bounds__(256) void k_merge_init(const float* __restrict__ x,
                                                    const int* __restrict__ edge_idx,
                                                    float* __restrict__ out) {
    const int row = (int)blockIdx.x;                 // 0 .. B*OUTT-1
    const int b = row / OUTT, t = row % OUTT;
    int src_t;
    if (t < HT - RR) {                               // unmerged (sorted order r:)
        src_t = 2 * edge_idx[(size_t)b * HT + RR + t];
    } else {                                         // dst rows = odd tokens
        src_t = 2 * (t - (HT - RR)) + 1;
    }
    const float* xrow = x + ((size_t)b * TT + src_t) * CC;
    float* orow = out + (size_t)row * CC;
    for (int c = (int)threadIdx.x; c < CC; c += 256) orow[c] = xrow[c];
}

__global__ __launch_bounds__(256) void k_size_init(float* __restrict__ sizes) {
    int i = (int)(blockIdx.x * blockDim.x + threadIdx.x);
    if (i < BB * OUTT) sizes[i] = 1.0f;
}

// ---------------------------------------------------------------------------
// 5) Scatter-add merged src rows into dst rows + bump sizes (float atomics).
// ---------------------------------------------------------------------------
__global__ __launch_bounds__(256) void k_scatter(const float* __restrict__ x,
                                                 const int* __restrict__ edge_idx,
                                                 const int* __restrict__ node_idx,
                                                 float* __restrict__ out,
                                                 float* __restrict__ sizes) {
    const int s = (int)blockIdx.x;                   // 0 .. B*R-1
    const int b = s / RR, si = s % RR;
    const int src_i = edge_idx[(size_t)b * HT + si];
    const int dst_j = node_idx[(size_t)b * HT + src_i];
    const float* xrow = x + ((size_t)b * TT + 2 * src_i) * CC;
    float* orow = out + ((size_t)b * OUTT + (HT - RR) + dst_j) * CC;
    for (int c = (int)threadIdx.x; c < CC; c += 256) atomicAdd(&orow[c], xrow[c]);
    if (threadIdx.x == 0) atomicAdd(&sizes[(size_t)b * OUTT + (HT - RR) + dst_j], 1.0f);
}

// ---------------------------------------------------------------------------
// 6) Weighted average: divide every output row by its size.
// ---------------------------------------------------------------------------
__global__ __launch_bounds__(256) void k_divide(float* __restrict__ out,
                                                const float* __restrict__ sizes) {
    const int row = (int)blockIdx.x;
    const float inv = 1.0f / sizes[row];
    float* orow = out + (size_t)row * CC;
    for (int c = (int)threadIdx.x; c < CC; c += 256) orow[c] *= inv;
}

// ---------------------------------------------------------------------------
extern "C" void kernel_launch(void* const* d_in, const int* in_sizes, int n_in,
                              void* d_out, int out_size, void* d_ws, size_t ws_size,
                              hipStream_t stream) {
    const float* x      = (const float*)d_in[0];
    const float* metric = (const float*)d_in[1];
    // d_in[2] holds r (==1024); shapes are compile-time constants here.

    // Workspace layout (~640 KB total).
    float* inv_norm = (float*)d_ws;                       // B*T
    float* node_max = inv_norm + (size_t)BB * TT;         // B*HT
    int*   node_idx = (int*)(node_max + (size_t)BB * HT); // B*HT
    int*   edge_idx = node_idx + (size_t)BB * HT;         // B*HT

    float* out   = (float*)d_out;                         // B*OUTT*C
    float* sizes = out + (size_t)BB * OUTT * CC;          // B*OUTT

    // 1) row norms of metric
    k_inv_norm<<<dim3((BB * TT) / 8), dim3(256), 0, stream>>>(metric, inv_norm);

    // 2) fused f32-WMMA scores + row argmax
    k_scores_argmax<<<dim3(HT / 64, BB), dim3(128), 0, stream>>>(metric, inv_norm,
                                                                 node_max, node_idx);

    // 3) stable descending argsort of node_max
    k_rank<<<dim3(BB), dim3(256), 0, stream>>>(node_max, edge_idx);

    // 4) init output rows + sizes
    k_merge_init<<<dim3(BB * OUTT), dim3(256), 0, stream>>>(x, edge_idx, out);
    k_size_init<<<dim3((BB * OUTT + 255) / 256), dim3(256), 0, stream>>>(sizes);

    // 5) scatter-add merged tokens
    k_scatter<<<dim3(BB * RR), dim3(256), 0, stream>>>(x, edge_idx, node_idx, out, sizes);

    // 6) weighted average
    k_divide<<<dim3(BB * OUTT), dim3(256), 0, stream>>>(out, sizes);
}